// EEG_Deformer_44092134261231
// MI455X (gfx1250) — compile-verified
//
#include <hip/hip_runtime.h>
#include <cstdint>
#include <math.h>

// EEG-Deformer deformable-offset gather, MI455X (gfx1250).
// out[b,c,t] = sum_k x_ext[ floor( tanh(conv5(x)[k,t]+b[k])*5 + t + (k-2) + 7 ) ]
// Memory-bound (~64MB @ 23.3TB/s). Strategy: async DMA each row into LDS once,
// do conv + v_tanh_f32 + LDS gathers, stream results out with NT stores.

#define K_TAPS 5
#define OFF_L  5.0f
#define T_LEN  4000
#define PAD    7                    // head_n = tail_n = K//2 + L = 7
#define EXT    (T_LEN + 2 * PAD)    // 4014 = len(x_ext)

#if __has_builtin(__builtin_amdgcn_tanhf)
  #define FAST_TANH(v) __builtin_amdgcn_tanhf(v)
#elif __has_builtin(__builtin_amdgcn_tanh_f32)
  #define FAST_TANH(v) __builtin_amdgcn_tanh_f32(v)
#else
  #define FAST_TANH(v) tanhf(v)
#endif

// Low 32 bits of a flat LDS pointer == LDS byte offset (aperture in high bits).
__device__ __forceinline__ uint32_t lds_off(const void* p) {
    return (uint32_t)(uintptr_t)p;
}

// CDNA5 async global->LDS DMA (GV mode, tracked by ASYNCcnt).
__device__ __forceinline__ void async_ld_b128(uint32_t lds, const float* g) {
    asm volatile("global_load_async_to_lds_b128 %0, %1, off"
                 :: "v"(lds), "v"(g) : "memory");
}
__device__ __forceinline__ void async_ld_b32(uint32_t lds, const float* g) {
    asm volatile("global_load_async_to_lds_b32 %0, %1, off"
                 :: "v"(lds), "v"(g) : "memory");
}
__device__ __forceinline__ void wait_async0() {
    asm volatile("s_wait_asynccnt 0x0" ::: "memory");
}

__global__ __launch_bounds__(256)
void eeg_deformer_kernel(const float* __restrict__ x,
                         const float* __restrict__ w_off,
                         const float* __restrict__ b_off,
                         float* __restrict__ out,
                         int rows)
{
    // s_x = s_mem+1 puts x[j] (stored at s_x[j+7]) at LDS byte base+4+(j+7)*4
    // = base+32+4j -> 16B-aligned whenever j%4==0, matching b128 chunks.
    __shared__ __attribute__((aligned(16))) float s_mem[EXT + 1];
    __shared__ float s_w[K_TAPS * K_TAPS];
    __shared__ float s_b[K_TAPS];
    float* s_x = s_mem + 1;   // s_x[i] == x_ext[i], i in [0, EXT)

    const int row = blockIdx.x;            // row = b*C + c
    if (row >= rows) return;
    const int tid = threadIdx.x;
    const float* xr = x + (size_t)row * T_LEN;

    // ---- stage x_ext row into LDS via async DMA ----
    // middle: x[0..T) -> s_x[7 .. T+7), 16B granules
    for (int c4 = tid; c4 < T_LEN / 4; c4 += 256) {
        const int j = c4 * 4;
        async_ld_b128(lds_off(&s_x[j + PAD]), xr + j);
    }
    if (tid < PAD) {
        // head: s_x[i]       = x[T-7+i]
        async_ld_b32(lds_off(&s_x[tid]), xr + (T_LEN - PAD) + tid);
        // tail: s_x[T+7+i]   = x[i]
        async_ld_b32(lds_off(&s_x[T_LEN + PAD + tid]), xr + tid);
    }
    if (tid < K_TAPS * K_TAPS) s_w[tid] = w_off[tid];   // w_off[k,0,0,j] -> k*5+j
    if (tid < K_TAPS)          s_b[tid] = b_off[tid];

    wait_async0();        // ASYNCcnt==0: our DMA landed in LDS
    __syncthreads();      // DScnt + barrier: whole tile visible to all waves

    // ---- compute ----
    for (int t = tid; t < T_LEN; t += 256) {
        // zero-padded 5-tap window of x around t (conv uses zero pad, not wrap)
        float win[K_TAPS];
        win[0] = (t >= 2)          ? s_x[t + PAD - 2] : 0.0f;
        win[1] = (t >= 1)          ? s_x[t + PAD - 1] : 0.0f;
        win[2] =                     s_x[t + PAD];
        win[3] = (t <= T_LEN - 2)  ? s_x[t + PAD + 1] : 0.0f;
        win[4] = (t <= T_LEN - 3)  ? s_x[t + PAD + 2] : 0.0f;

        float acc = 0.0f;
        #pragma unroll
        for (int k = 0; k < K_TAPS; ++k) {
            float conv = s_b[k];
            #pragma unroll
            for (int j = 0; j < K_TAPS; ++j)
                conv = fmaf(win[j], s_w[k * K_TAPS + j], conv);
            const float th   = FAST_TANH(conv);
            const float pos  = fmaf(th, OFF_L, (float)(t + (k - 2) + PAD));
            const float left = floorf(pos);
            int idx = (int)left;
            idx = idx < 0 ? 0 : (idx > EXT - 1 ? EXT - 1 : idx);
            const float v  = s_x[idx];            // LDS gather
            const float lr = pos - left;          // faithful to reference:
            const float rr = left + 1.0f - pos;   // lr*v + rr*v (== v up to rounding)
            acc += lr * v + rr * v;
        }
        __builtin_nontemporal_store(acc, out + (size_t)row * T_LEN + t);
    }
}

extern "C" void kernel_launch(void* const* d_in, const int* in_sizes, int n_in,
                              void* d_out, int out_size, void* d_ws, size_t ws_size,
                              hipStream_t stream) {
    (void)n_in; (void)d_ws; (void)ws_size; (void)out_size;
    const float* x     = (const float*)d_in[0];   // (B,1,C,T) f32
    const float* w_off = (const float*)d_in[1];   // (K,1,1,K) f32
    const float* b_off = (const float*)d_in[2];   // (K,)      f32
    float* out = (float*)d_out;                   // (B,1,C,T) f32

    const int rows = in_sizes[0] / T_LEN;         // B*C = 2048
    eeg_deformer_kernel<<<rows, 256, 0, stream>>>(x, w_off, b_off, out, rows);
}